// SegSAGE_52974126629691
// MI455X (gfx1250) — compile-verified
//
#include <hip/hip_runtime.h>
#include <hip/hip_bf16.h>
#include <stdint.h>

// Problem constants (match the reference)
#define NN 100000      // nodes
#define NE 1600000     // edges (before self loops)
#define FD 128         // feature / hidden dim
#define NC 40          // classes
#define NP 48          // classes padded to 3x16 WMMA tiles

typedef __attribute__((ext_vector_type(2))) float v2f;
typedef __attribute__((ext_vector_type(8))) float v8f;

static __device__ __forceinline__ void atomAddF(float* p, float v) {
    __hip_atomic_fetch_add(p, v, __ATOMIC_RELAXED, __HIP_MEMORY_SCOPE_AGENT);
}

// ---------------- degree: deg[v] = 1 (self loop) + #incoming; invert in place ----
__global__ void k_deg_init(float* __restrict__ deg) {
    int i = blockIdx.x * blockDim.x + threadIdx.x;
    if (i < NN) deg[i] = 1.0f;
}
__global__ void k_deg_count(const int* __restrict__ dst, float* __restrict__ deg) {
    int e = blockIdx.x * blockDim.x + threadIdx.x;
    if (e < NE) atomAddF(&deg[dst[e]], 1.0f);
}
__global__ void k_deg_inv(float* __restrict__ deg) {
    int i = blockIdx.x * blockDim.x + threadIdx.x;
    if (i < NN) deg[i] = 1.0f / deg[i];
}

// ---------------- weight repack into WMMA-B fragment order ----------------
// P[2*((t*2+h)*FD + n)] = W[(4t+2h)  ][n]
// P[2*((t*2+h)*FD + n)+1] = W[(4t+2h+1)][n]
// so lane (h, r) of tile n0 at k-step t loads one contiguous float2.
__global__ void k_pack_w(const float* __restrict__ W, float* __restrict__ P) {
    int j = blockIdx.x * blockDim.x + threadIdx.x;   // [0, 32*2*FD)
    if (j >= 32 * 2 * FD) return;
    int n = j & (FD - 1);
    int h = (j >> 7) & 1;
    int t = j >> 8;
    P[2 * j + 0] = W[(4 * t + 2 * h) * FD + n];
    P[2 * j + 1] = W[(4 * t + 2 * h + 1) * FD + n];
}
// same, for W_score with zero-padding 40 -> 48 columns
__global__ void k_pack_wsc(const float* __restrict__ W, float* __restrict__ P) {
    int j = blockIdx.x * blockDim.x + threadIdx.x;   // [0, 32*2*NP)
    if (j >= 32 * 2 * NP) return;
    int n = j % NP;
    int h = (j / NP) & 1;
    int t = j / (2 * NP);
    float x = 0.0f, y = 0.0f;
    if (n < NC) {
        x = W[(4 * t + 2 * h) * NC + n];
        y = W[(4 * t + 2 * h + 1) * NC + n];
    }
    P[2 * j + 0] = x;
    P[2 * j + 1] = y;
}
__global__ void k_pack_bsc(const float* __restrict__ b, float* __restrict__ P) {
    int j = blockIdx.x * blockDim.x + threadIdx.x;
    if (j < NP) P[j] = (j < NC) ? b[j] : 0.0f;
}

// ---------------- aggregate init: agg[v] = h[v] (self-loop term) ----------------
__global__ void k_agg_init(const float4* __restrict__ h, float4* __restrict__ agg) {
    int i = blockIdx.x * blockDim.x + threadIdx.x;
    if (i < NN * (FD / 4)) agg[i] = h[i];
}

// ---------------- edge scatter: agg[dst] += h[src] ----------------
// One wave per edge; edge index forced into an SGPR so src/dst become s_load.
// Lane carries 4 floats: one b128 gather + 4 f32 atomics (L2-resident).
__global__ __launch_bounds__(256) void k_edge_scatter(const float* __restrict__ h,
                                                      const int* __restrict__ src,
                                                      const int* __restrict__ dst,
                                                      float* __restrict__ agg) {
    const int e = __builtin_amdgcn_readfirstlane(blockIdx.x * 8 + (threadIdx.x >> 5));
    const int lane = threadIdx.x & 31;
    const int s = src[e];
    const int d = dst[e];
    float4 v = ((const float4*)h)[(size_t)s * (FD / 4) + lane];
    float* o = agg + (size_t)d * FD + lane * 4;
    atomAddF(o + 0, v.x);
    atomAddF(o + 1, v.y);
    atomAddF(o + 2, v.z);
    atomAddF(o + 3, v.w);
}

// ---------------- SAGE GEMM ----------------
// out = Aself@Wself + (Aneigh*deg_inv[row])@Wneigh + b, optional LeakyReLU.
// One wave computes TWO 16x16 tiles (rows m0..m0+15 and m0+16..m0+31) sharing
// the same B fragments; 8 waves/block cover all 128 output cols. Grid = NN/32.
// f32 WMMA 16x16x4 layouts (CDNA5 ISA 7.12.2):
//   A: lane (h, r): {A[m0+r][k0+2h], A[m0+r][k0+2h+1]}   -> one b64 load
//   B: lane (h, r): {W[k0+2h][n0+r], W[k0+2h+1][n0+r]}   -> one b64 load (packed)
//   C/D: vgpr i -> row m0 + i + 8*h, col n0 + r
template <bool LEAKY>
__global__ __launch_bounds__(256) void k_sage_gemm(
    const float* __restrict__ Aself, const float* __restrict__ Aneigh,
    const float* __restrict__ dinv,
    const v2f* __restrict__ PBs, const v2f* __restrict__ PBn,
    const float* __restrict__ bias, float* __restrict__ out) {
    const int m0   = blockIdx.x * 32;
    const int wave = threadIdx.x >> 5;
    const int lane = threadIdx.x & 31;
    const int n0   = wave * 16;
    const int half = lane >> 4;
    const int r    = lane & 15;
    const int row0 = m0 + r;
    const int row1 = m0 + 16 + r;

    const float ds0 = dinv[row0];
    const float ds1 = dinv[row1];
    const float* as0 = Aself  + (size_t)row0 * FD;
    const float* an0 = Aneigh + (size_t)row0 * FD;
    const float* as1 = Aself  + (size_t)row1 * FD;
    const float* an1 = Aneigh + (size_t)row1 * FD;
    const int bbase = half * FD + n0 + r;

    v8f c0 = {}, c1 = {};
    #pragma unroll 8
    for (int t = 0; t < FD / 4; ++t) {
        const int ka = 4 * t + 2 * half;
        const v2f bs = PBs[2 * t * FD + bbase];
        const v2f bn = PBn[2 * t * FD + bbase];

        v2f a0s = *(const v2f*)(as0 + ka);
        c0 = __builtin_amdgcn_wmma_f32_16x16x4_f32(false, a0s, false, bs,
                                                   (short)0, c0, false, false);
        v2f a1s = *(const v2f*)(as1 + ka);
        c1 = __builtin_amdgcn_wmma_f32_16x16x4_f32(false, a1s, false, bs,
                                                   (short)0, c1, false, false);
        v2f a0n = *(const v2f*)(an0 + ka);
        a0n.x *= ds0; a0n.y *= ds0;
        c0 = __builtin_amdgcn_wmma_f32_16x16x4_f32(false, a0n, false, bn,
                                                   (short)0, c0, false, false);
        v2f a1n = *(const v2f*)(an1 + ka);
        a1n.x *= ds1; a1n.y *= ds1;
        c1 = __builtin_amdgcn_wmma_f32_16x16x4_f32(false, a1n, false, bn,
                                                   (short)0, c1, false, false);
    }

    const float bb = bias[n0 + r];
    #pragma unroll
    for (int i = 0; i < 8; ++i) {
        float v0 = c0[i] + bb;
        float v1 = c1[i] + bb;
        if (LEAKY) {
            v0 = (v0 >= 0.0f) ? v0 : 0.01f * v0;
            v1 = (v1 >= 0.0f) ? v1 : 0.01f * v1;
        }
        out[(size_t)(m0 + i + 8 * half) * FD + (n0 + r)]      = v0;
        out[(size_t)(m0 + 16 + i + 8 * half) * FD + (n0 + r)] = v1;
    }
}

// ---------------- score GEMM: out[N x 40] = H @ Wsc + bsc ----------------
// Pre-padded/packed weights (NP=48 cols) -> branch-free inner loop.
// 3 waves/block (one per 16-col tile), 2 M tiles per wave. Grid = NN/32.
__global__ __launch_bounds__(96) void k_score_gemm(
    const float* __restrict__ H, const v2f* __restrict__ PW,
    const float* __restrict__ pb, float* __restrict__ out) {
    const int m0   = blockIdx.x * 32;
    const int wave = threadIdx.x >> 5;
    const int lane = threadIdx.x & 31;
    const int n0   = wave * 16;
    const int half = lane >> 4;
    const int r    = lane & 15;
    const int col  = n0 + r;
    const float* a0 = H + (size_t)(m0 + r) * FD;
    const float* a1 = H + (size_t)(m0 + 16 + r) * FD;
    const int bbase = half * NP + col;

    v8f c0 = {}, c1 = {};
    #pragma unroll 8
    for (int t = 0; t < FD / 4; ++t) {
        const int ka = 4 * t + 2 * half;
        const v2f b = PW[2 * t * NP + bbase];
        v2f x0 = *(const v2f*)(a0 + ka);
        c0 = __builtin_amdgcn_wmma_f32_16x16x4_f32(false, x0, false, b,
                                                   (short)0, c0, false, false);
        v2f x1 = *(const v2f*)(a1 + ka);
        c1 = __builtin_amdgcn_wmma_f32_16x16x4_f32(false, x1, false, b,
                                                   (short)0, c1, false, false);
    }
    if (col < NC) {
        const float bb = pb[col];
        #pragma unroll
        for (int i = 0; i < 8; ++i) {
            out[(size_t)(m0 + i + 8 * half) * NC + col]      = c0[i] + bb;
            out[(size_t)(m0 + 16 + i + 8 * half) * NC + col] = c1[i] + bb;
        }
    }
}

extern "C" void kernel_launch(void* const* d_in, const int* in_sizes, int n_in,
                              void* d_out, int out_size, void* d_ws, size_t ws_size,
                              hipStream_t stream) {
    (void)in_sizes; (void)n_in; (void)out_size; (void)ws_size;
    const float* feat = (const float*)d_in[0];
    const int*   src  = (const int*)  d_in[1];
    const int*   dst  = (const int*)  d_in[2];
    const float* W1s  = (const float*)d_in[3];
    const float* W1n  = (const float*)d_in[4];
    const float* b1   = (const float*)d_in[5];
    const float* W2s  = (const float*)d_in[6];
    const float* W2n  = (const float*)d_in[7];
    const float* b2   = (const float*)d_in[8];
    const float* Wsc  = (const float*)d_in[9];
    const float* bsc  = (const float*)d_in[10];
    float* out = (float*)d_out;

    // workspace carving
    char* p = (char*)d_ws;
    const size_t MAT  = (size_t)NN * FD * sizeof(float);        // 51.2 MB
    const size_t WPK  = (size_t)FD * FD * sizeof(float);        // 64 KB packed weight
    const size_t WSCP = (size_t)FD * NP * sizeof(float);        // 24 KB packed score w
    float* dinv = (float*)p;  p += ((size_t)NN * sizeof(float) + 511) & ~(size_t)511;
    float* agg  = (float*)p;  p += MAT;
    float* h1   = (float*)p;  p += MAT;
    float* h2   = (float*)p;  p += MAT;
    float* pw1s = (float*)p;  p += WPK;
    float* pw1n = (float*)p;  p += WPK;
    float* pw2s = (float*)p;  p += WPK;
    float* pw2n = (float*)p;  p += WPK;
    float* pwsc = (float*)p;  p += WSCP;
    float* pbsc = (float*)p;

    const int BT = 256;
    const int gN   = (NN + BT - 1) / BT;
    const int gE   = (NE + BT - 1) / BT;
    const int gCpy = (NN * (FD / 4) + BT - 1) / BT;
    const int gPk  = (32 * 2 * FD + BT - 1) / BT;
    const int gPkS = (32 * 2 * NP + BT - 1) / BT;
    const int gScat = NE / 8;        // one wave per edge, 8 waves/block
    const int gM   = NN / 32;        // 3125 row blocks (32 rows each)

    // weight repack (independent of graph work)
    k_pack_w  <<<gPk,  BT, 0, stream>>>(W1s, pw1s);
    k_pack_w  <<<gPk,  BT, 0, stream>>>(W1n, pw1n);
    k_pack_w  <<<gPk,  BT, 0, stream>>>(W2s, pw2s);
    k_pack_w  <<<gPk,  BT, 0, stream>>>(W2n, pw2n);
    k_pack_wsc<<<gPkS, BT, 0, stream>>>(Wsc, pwsc);
    k_pack_bsc<<<1,    64, 0, stream>>>(bsc, pbsc);

    // degrees -> deg_inv
    k_deg_init <<<gN, BT, 0, stream>>>(dinv);
    k_deg_count<<<gE, BT, 0, stream>>>(dst, dinv);
    k_deg_inv  <<<gN, BT, 0, stream>>>(dinv);

    // layer 1: aggregate feat, GEMM + bias + LeakyReLU
    k_agg_init    <<<gCpy,  BT, 0, stream>>>((const float4*)feat, (float4*)agg);
    k_edge_scatter<<<gScat, BT, 0, stream>>>(feat, src, dst, agg);
    k_sage_gemm<true><<<gM, 256, 0, stream>>>(feat, agg, dinv,
                                              (const v2f*)pw1s, (const v2f*)pw1n, b1, h1);

    // layer 2: aggregate h1, GEMM + bias (no activation)
    k_agg_init    <<<gCpy,  BT, 0, stream>>>((const float4*)h1, (float4*)agg);
    k_edge_scatter<<<gScat, BT, 0, stream>>>(h1, src, dst, agg);
    k_sage_gemm<false><<<gM, 256, 0, stream>>>(h1, agg, dinv,
                                               (const v2f*)pw2s, (const v2f*)pw2n, b2, h2);

    // score head
    k_score_gemm<<<gM, 96, 0, stream>>>(h2, (const v2f*)pwsc, pbsc, out);
}